// QuantumAttention_65481071410280
// MI455X (gfx1250) — compile-verified
//
#include <hip/hip_runtime.h>

// QuantumAttention, fully fused for gfx1250 (MI455X).
//
// Key math reduction: the 8-wire quantum circuit (RX layers -> CNOT ring) on a
// product state reduces to z_w = cos(angle_w + theta_w) followed by prefix
// products: out[w>=1] = z_0*...*z_w, out[0] = z_1*...*z_7  (CNOTs act as prefix
// XOR on independent bits; E[(-1)^XOR] = product of per-bit expectations).
//
// All projections are done with V_WMMA_F32_16X16X4_F32 (exact fp32, matches the
// fp32 reference). Each wave processes a 16-row tile = 2 batch elements.

typedef float v2f __attribute__((ext_vector_type(2)));
typedef float v8f __attribute__((ext_vector_type(8)));

constexpr int NBATCH     = 16384;
constexpr int TILES      = NBATCH * 8 / 16;   // 8192 16-row tiles
constexpr int WPW        = 8;                 // waves per workgroup
constexpr int LDS_STRIDE = 20;                // floats; conflict-free & 8B-aligned

__device__ __forceinline__ void lds_fence() {
  // per-wave ordering of the C-layout -> A-layout LDS round trip
  asm volatile("s_wait_dscnt 0x0" ::: "memory");
}

__device__ __forceinline__ v2f make2(float a, float b) {
  v2f r; r.x = a; r.y = b; return r;
}

__device__ __forceinline__ v8f splat8(float f) {
  v8f r;
#pragma unroll
  for (int i = 0; i < 8; ++i) r[i] = f;
  return r;
}

__device__ __forceinline__ v8f wmma4(v2f a, v2f b, v8f c) {
  // D(16x16,f32) = A(16x4,f32) x B(4x16,f32) + C   -- v_wmma_f32_16x16x4_f32
  return __builtin_amdgcn_wmma_f32_16x16x4_f32(
      false, a, false, b, (short)0, c, false, false);
}

// Relayout a 16x16 f32 tile from WMMA C/D layout (lane=column, vgpr=row) to
// WMMA A layout (lane=row, K split across lane halves), keeping K=0..7 only.
__device__ __forceinline__ void c_to_a(float* tb, const v8f& d,
                                       int lane, v2f& a_lo, v2f& a_hi) {
  const int col  = lane & 15;
  const int half = lane >> 4;
  const int kb   = half * 2;
#pragma unroll
  for (int r = 0; r < 8; ++r) tb[(half * 8 + r) * LDS_STRIDE + col] = d[r];
  lds_fence();
  a_lo = *(const v2f*)(tb + col * LDS_STRIDE + kb);
  a_hi = *(const v2f*)(tb + col * LDS_STRIDE + kb + 4);
  lds_fence();
}

__global__ __launch_bounds__(WPW * 32) void qattn_kernel(
    const float* __restrict__ x,    // (16384, 8, 8)
    const float* __restrict__ Wi,   // in_proj_w   (24, 8)
    const float* __restrict__ Bi,   // in_proj_b   (24)
    const float* __restrict__ Wo,   // out_proj_w  (8, 8)
    const float* __restrict__ Bo,   // out_proj_b  (8)
    const float* __restrict__ Th,   // theta       (8, 8)
    const float* __restrict__ Wc,   // combine_w   (8, 8)
    const float* __restrict__ Bc,   // combine_b   (8)
    float* __restrict__ out)        // (16384, 8, 8)
{
  __shared__ __align__(16) float lds[WPW][16 * LDS_STRIDE];
  const int lane = threadIdx.x & 31;
  const int wave = threadIdx.x >> 5;
  const int tile = blockIdx.x * WPW + wave;          // 16-row tile index
  float* tb = lds[wave];

  const int col  = lane & 15;   // column (C-layout) / row M (A-layout)
  const int half = lane >> 4;   // 0: rows 0-7 / K 0-1 ; 1: rows 8-15 / K 2-3
  const int kb   = half * 2;    // K base within a K=4 block
  const int w7   = lane & 7;    // wire / head index within 8-lane segment

  // ---------- load X tile directly in A layout ----------
  const float* xr = x + (size_t)(tile * 16 + col) * 8;
  const v2f ax_lo = *(const v2f*)(xr + kb);        // K = kb, kb+1
  const v2f ax_hi = *(const v2f*)(xr + kb + 4);    // K = kb+4, kb+5

  // ---------- in_proj: qkv = X @ Wi^T + Bi  (N tiles: cols 0-15 and 16-23) ----------
  const float* wr0 = Wi + col * 8;                 // B[k,n] = Wi[n,k]
  const v2f bi0_lo = make2(wr0[kb],     wr0[kb + 1]);
  const v2f bi0_hi = make2(wr0[kb + 4], wr0[kb + 5]);
  v2f bi1_lo = make2(0.f, 0.f), bi1_hi = make2(0.f, 0.f);
  float bias1 = 0.f;
  if (col + 16 < 24) {
    const float* wr1 = Wi + (col + 16) * 8;
    bi1_lo = make2(wr1[kb],     wr1[kb + 1]);
    bi1_hi = make2(wr1[kb + 4], wr1[kb + 5]);
    bias1  = Bi[col + 16];
  }
  v8f d0 = splat8(Bi[col]);                        // bias preloaded into C
  v8f d1 = splat8(bias1);
  d0 = wmma4(ax_lo, bi0_lo, d0); d0 = wmma4(ax_hi, bi0_hi, d0);  // qkv cols 0-15
  d1 = wmma4(ax_lo, bi1_lo, d1); d1 = wmma4(ax_hi, bi1_hi, d1);  // qkv cols 16-23

  // ---------- attention (D_K = 1): lane h owns head h of its batch ----------
  // d0 lane h: q[:,h]; d0 lane h+8: k[:,h]; d1 lane h: v[:,h]
  const int ksrc = (lane & 16) | (8 + w7);
  float q[8], k[8], v[8];
#pragma unroll
  for (int r = 0; r < 8; ++r) {
    q[r] = d0[r];
    k[r] = __shfl(d0[r], ksrc, 32);
    v[r] = d1[r];
  }
  float ctx[8];
#pragma unroll
  for (int i = 0; i < 8; ++i) {
    float s[8];
    float mx = q[i] * k[0]; s[0] = mx;
#pragma unroll
    for (int j = 1; j < 8; ++j) { s[j] = q[i] * k[j]; mx = fmaxf(mx, s[j]); }
    float sum = 0.f, acc = 0.f;
#pragma unroll
    for (int j = 0; j < 8; ++j) {
      float e = __expf(s[j] - mx);
      sum += e;
      acc = fmaf(e, v[j], acc);
    }
    ctx[i] = acc / sum;   // ctx_mat[:, h] in C layout (sum >= 1, safe)
  }

  // ---------- out_proj: attn_out = ctx @ Wo^T + Bo ----------
  v2f a_lo, a_hi;
  {
    v8f cvec;                       // pack ctx into a v8f (C layout)
#pragma unroll
    for (int r = 0; r < 8; ++r) cvec[r] = ctx[r];
    c_to_a(tb, cvec, lane, a_lo, a_hi);
  }
  v2f bo_lo = make2(0.f, 0.f), bo_hi = make2(0.f, 0.f);
  float biaso = 0.f;
  if (col < 8) {
    const float* wr = Wo + col * 8;
    bo_lo = make2(wr[kb],     wr[kb + 1]);
    bo_hi = make2(wr[kb + 4], wr[kb + 5]);
    biaso = Bo[col];
  }
  v8f ao = splat8(biaso);
  ao = wmma4(a_lo, bo_lo, ao); ao = wmma4(a_hi, bo_hi, ao);   // attn_out, C layout

  // ---------- quantum layer (analytic): cos + 8-lane prefix product ----------
  float z[8], zp[8];
#pragma unroll
  for (int r = 0; r < 8; ++r) {
    z[r]  = __cosf(ao[r] + Th[r * 8 + w7]);       // row r -> seq s=r -> theta[s][w]
    zp[r] = (w7 == 0) ? 1.f : z[r];               // z' (wire0 replaced by 1)
  }
#pragma unroll
  for (int off = 1; off < 8; off <<= 1) {
#pragma unroll
    for (int r = 0; r < 8; ++r) {
      float t = __shfl(zp[r], lane - off, 32);
      zp[r] = (w7 >= off) ? zp[r] * t : zp[r];    // inclusive scan of z'
    }
  }
  const int segb = lane & 24;
  v8f qo;
#pragma unroll
  for (int r = 0; r < 8; ++r) {
    float z0  = __shfl(z[r],  segb,     32);      // z_0
    float tot = __shfl(zp[r], segb | 7, 32);      // z_1*...*z_7
    qo[r] = (w7 == 0) ? tot : z0 * zp[r];         // out[w>=1] = z0*z1*..*zw
  }

  // ---------- combine (shared weights for both applications) ----------
  v2f bc_lo = make2(0.f, 0.f), bc_hi = make2(0.f, 0.f);
  float biasc = 0.f;
  if (col < 8) {
    const float* wr = Wc + col * 8;
    bc_lo = make2(wr[kb],     wr[kb + 1]);
    bc_hi = make2(wr[kb + 4], wr[kb + 5]);
    biasc = Bc[col];
  }

  // q1 = quantum_out @ Wc^T + Bc
  c_to_a(tb, qo, lane, a_lo, a_hi);
  v8f q1 = splat8(biasc);
  q1 = wmma4(a_lo, bc_lo, q1); q1 = wmma4(a_hi, bc_hi, q1);

  // final = (attn_out + q1) @ Wc^T + Bc
  v8f y = ao + q1;
  c_to_a(tb, y, lane, a_lo, a_hi);
  v8f dout = splat8(biasc);
  dout = wmma4(a_lo, bc_lo, dout); dout = wmma4(a_hi, bc_hi, dout);

  // ---------- store (predicated AFTER the last WMMA) ----------
  if (col < 8) {
    float* orow = out + (size_t)(tile * 16 + half * 8) * 8 + col;
#pragma unroll
    for (int r = 0; r < 8; ++r) orow[r * 8] = dout[r];
  }
}

extern "C" void kernel_launch(void* const* d_in, const int* in_sizes, int n_in,
                              void* d_out, int out_size, void* d_ws, size_t ws_size,
                              hipStream_t stream) {
  (void)in_sizes; (void)n_in; (void)out_size; (void)d_ws; (void)ws_size;
  const float* x  = (const float*)d_in[0];
  const float* Wi = (const float*)d_in[1];
  const float* Bi = (const float*)d_in[2];
  const float* Wo = (const float*)d_in[3];
  const float* Bo = (const float*)d_in[4];
  const float* Th = (const float*)d_in[5];
  const float* Wc = (const float*)d_in[6];
  const float* Bc = (const float*)d_in[7];
  float* out = (float*)d_out;

  qattn_kernel<<<dim3(TILES / WPW), dim3(WPW * 32), 0, stream>>>(
      x, Wi, Bi, Wo, Bo, Th, Wc, Bc, out);
}